// Raster_12996571037982
// MI455X (gfx1250) — compile-verified
//
#include <hip/hip_runtime.h>

#define NPT   65536
#define P     14
#define PP    (P * P)       // 196
#define PATCH (P * P * P)   // 2744

typedef float v2f __attribute__((ext_vector_type(2)));
typedef float v8f __attribute__((ext_vector_type(8)));

// Per-axis: lane L computes cdf at edge L; q_L = cdf_{L+1} - cdf_L via shfl_down.
// Must be called at full EXEC (uses cross-lane shuffle).
__device__ __forceinline__ float axis_q(float c, float s, float kf, bool valid,
                                        float& iminf) {
  const float inv_sp = 2.0f;   // 1 / 0.5 spacing
  const float sp     = 0.5f;
  iminf = floorf((c - 3.0f * s) * inv_sp);
  const float invs = 0.70710678118654752f / s;   // 1/(sqrt(2)*sigma)
  const float e  = (iminf + kf) * sp;
  const float cd = 0.5f * erff((e - c) * invs);  // one erf per lane per axis
  const float cd1 = __shfl_down(cd, 1, 32);      // cdf at edge L+1
  return valid ? (cd1 - cd) : 0.0f;
}

__global__ __launch_bounds__(256) void raster_wmma_kernel(
    const float* __restrict__ sigma,   // (NPT,3)
    const float* __restrict__ time,    // (NPT,)
    const float* __restrict__ charge,  // (NPT,)
    const float* __restrict__ tail,    // (NPT,3)
    float* __restrict__ out_rasters,   // (NPT,14,14,14)
    int*   __restrict__ out_offsets)   // (NPT,3)
{
  const int lane = threadIdx.x & 31;
  const int wave = threadIdx.x >> 5;
  const int n    = blockIdx.x * (blockDim.x >> 5) + wave;

  // centers: x <- tail[:,1], y <- tail[:,0], z <- time  (reference transform)
  const float cx = tail[3 * n + 1];
  const float cy = tail[3 * n + 0];
  const float cz = time[n];
  const float sx = sigma[3 * n + 0];
  const float sy = sigma[3 * n + 1];
  const float sz = sigma[3 * n + 2];
  const float q  = charge[n];

  const float kf     = (float)lane;
  const bool  kvalid = lane < P;

  float ixf, iyf, izf;
  const float qx = axis_q(cx, sx, kf, kvalid, ixf);
  const float qy = axis_q(cy, sy, kf, kvalid, iyf);
  const float qz = axis_q(cz, sz, kf, kvalid, izf);

  if (lane == 0) {
    out_offsets[3 * n + 0] = (int)ixf;
    out_offsets[3 * n + 1] = (int)iyf;
    out_offsets[3 * n + 2] = (int)izf;
  }

  // Rank-1 outer product qy (16x1) x qz (1x16) via V_WMMA_F32_16X16X4_F32.
  // A 16x4: VGPR0 = {K=0 | lanes 0-15, K=2 | lanes 16-31}; lanes >= 14 hold 0,
  // so a.x is the K=0 column and zeros elsewhere; a.y (K=1/K=3) zeroed.
  v2f a; a.x = qy; a.y = 0.0f;
  v2f b; b.x = qz; b.y = 0.0f;
  v8f cacc = {};
  v8f d = __builtin_amdgcn_wmma_f32_16x16x4_f32(
      false, a, false, b, (short)0, cacc, false, false);

  // Broadcast charge*qx[i] for all 14 x-slices at FULL exec (before masking).
  const float cq = q * qx;  // lane i holds charge*qx[i]
  float s[P];
#pragma unroll
  for (int i = 0; i < P; ++i) s[i] = __shfl(cq, i, 32);

  // D layout: VGPR r holds element (row = rowbase + r, col = lane&15),
  // rowbase = 8*(lane>=16). Single base address; all stores use immediate
  // offsets (i*196 + r*14)*4 <= 10920 bytes.
  const int  col     = lane & 15;
  const int  rowbase = (lane >> 4) << 3;
  const bool colok   = col < P;

  float* base = out_rasters + (size_t)n * PATCH + rowbase * P + col;

  // Region A: rows rowbase+0..rowbase+5 are always < 14 -> mask = colok only.
  if (colok) {
#pragma unroll
    for (int r = 0; r < 6; ++r) {
#pragma unroll
      for (int i = 0; i < P; ++i) {
        __builtin_nontemporal_store(d[r] * s[i], base + i * PP + r * P);
      }
    }
  }
  // Region B: rows rowbase+6, rowbase+7 valid only when rowbase == 0.
  if (colok && rowbase == 0) {
#pragma unroll
    for (int r = 6; r < 8; ++r) {
#pragma unroll
      for (int i = 0; i < P; ++i) {
        __builtin_nontemporal_store(d[r] * s[i], base + i * PP + r * P);
      }
    }
  }
}

extern "C" void kernel_launch(void* const* d_in, const int* in_sizes, int n_in,
                              void* d_out, int out_size, void* d_ws, size_t ws_size,
                              hipStream_t stream) {
  const float* sigma  = (const float*)d_in[0];
  const float* time   = (const float*)d_in[1];
  const float* charge = (const float*)d_in[2];
  const float* tail   = (const float*)d_in[3];

  float* out_rasters = (float*)d_out;
  int*   out_offsets = (int*)(out_rasters + (size_t)NPT * PATCH);

  // 256 threads = 8 wave32 per block, one depo per wave.
  dim3 block(256);
  dim3 grid(NPT / 8);
  hipLaunchKernelGGL(raster_wmma_kernel, grid, block, 0, stream,
                     sigma, time, charge, tail, out_rasters, out_offsets);
}